// Up_Sampling_2199023255677
// MI455X (gfx1250) — compile-verified
//
#include <hip/hip_runtime.h>

typedef __attribute__((ext_vector_type(2))) float v2f;
typedef __attribute__((ext_vector_type(8))) float v8f;
typedef int v4i __attribute__((vector_size(4 * sizeof(int))));

#define DIM 512
#define N1C 1024
#define N2C 4096
#define BC  16
#define R1  (BC * N1C)   // 16384 rows for GEMM1
#define R2  (BC * N2C)   // 65536 rows for GEMM2

// GEMM tiling: block = 128 threads (4 waves); block tile 128(M) x 64(N), K-chunk 32.
// Each wave owns 32 M-rows (two 16-row WMMA subtiles) and all 64 N columns:
// per K-step of 4 -> 8 WMMAs sharing 4 B operand pairs (B reuse x2).
#define TM  128
#define TN  64
#define TK  32
#define LDA 36           // padded A row stride (16B aligned, conflict-free)
#define LDB 68           // padded B row stride (16B aligned)
#define NTHR 128

#if __has_builtin(__builtin_amdgcn_global_load_async_to_lds_b128) && \
    __has_builtin(__builtin_amdgcn_s_wait_asynccnt)
#define ASYNC_COPY 1
#else
#define ASYNC_COPY 0
#endif

// One 16-byte async global->LDS transfer per lane (ASYNCcnt tracked).
__device__ __forceinline__ void copy_b128(const float* g, float* l)
{
#if ASYNC_COPY
    __builtin_amdgcn_global_load_async_to_lds_b128((v4i*)g, (v4i*)l, 0, 0);
#else
    *reinterpret_cast<float4*>(l) = *reinterpret_cast<const float4*>(g);
#endif
}

__device__ __forceinline__ void wait_async_copies()
{
#if ASYNC_COPY
    __builtin_amdgcn_s_wait_asynccnt(0);
#endif
}

// ---------------------------------------------------------------------------
// GEMM + bias: Y[M,512] = A[M,K] @ W[K,512] + bias
// Double-buffered LDS staging via async global->LDS; v_wmma_f32_16x16x4_f32.
// ---------------------------------------------------------------------------
template <int K>
__global__ __launch_bounds__(NTHR) void gemm_bias_wmma(
    const float* __restrict__ A, const float* __restrict__ W,
    const float* __restrict__ bias, float* __restrict__ Y)
{
    __shared__ __align__(16) float As[2][TM][LDA];
    __shared__ __align__(16) float Bs[2][TK][LDB];

    const int tid  = threadIdx.x;
    const int wave = tid >> 5;
    const int lane = tid & 31;
    const int half = lane >> 4;   // 0: lanes 0-15, 1: lanes 16-31
    const int l16  = lane & 15;
    const int m0   = blockIdx.x * TM;
    const int n0   = blockIdx.y * TN;

    v8f acc[2][4] = {};

    // Stage one K-chunk (A: 128x32, B: 32x64) into LDS buffer `buf`.
    auto stage = [&](int buf, int kc) {
        // A: 1024 float4 -> 8 per thread; 8 float4 per row of 32 floats
#pragma unroll
        for (int i = 0; i < 8; ++i) {
            const int idx = tid + i * NTHR;
            const int r = idx >> 3;
            const int c = (idx & 7) << 2;
            copy_b128(A + (size_t)(m0 + r) * K + kc + c, &As[buf][r][c]);
        }
        // B: 512 float4 -> 4 per thread; 16 float4 per row of 64 floats
#pragma unroll
        for (int i = 0; i < 4; ++i) {
            const int idx = tid + i * NTHR;
            const int r = idx >> 4;
            const int c = (idx & 15) << 2;
            copy_b128(W + (size_t)(kc + r) * DIM + n0 + c, &Bs[buf][r][c]);
        }
    };

    stage(0, 0);
    wait_async_copies();
    __syncthreads();

    const int nch = K / TK;
    for (int ch = 0; ch < nch; ++ch) {
        const int buf = ch & 1;
        if (ch + 1 < nch) stage(buf ^ 1, (ch + 1) * TK);

        // Two 16-row A subtiles per wave
        const float* Arow0 = &As[buf][wave * 32 + l16][0];
        const float* Arow1 = &As[buf][wave * 32 + 16 + l16][0];
#pragma unroll
        for (int kk = 0; kk < TK; kk += 4) {
            // A operands (16x4 f32): lanes 0-15 -> K=kk+{0,1}; lanes 16-31 -> K=kk+{2,3}
            v2f a0, a1;
            a0.x = Arow0[kk + half * 2 + 0];
            a0.y = Arow0[kk + half * 2 + 1];
            a1.x = Arow1[kk + half * 2 + 0];
            a1.y = Arow1[kk + half * 2 + 1];
#pragma unroll
            for (int j = 0; j < 4; ++j) {
                // B operand (4x16 f32): lane = N (mod 16), VGPR index = K in lane half
                v2f b;
                b.x = Bs[buf][kk + half * 2 + 0][j * 16 + l16];
                b.y = Bs[buf][kk + half * 2 + 1][j * 16 + l16];
                acc[0][j] = __builtin_amdgcn_wmma_f32_16x16x4_f32(
                    false, a0, false, b, (short)0, acc[0][j], false, false);
                acc[1][j] = __builtin_amdgcn_wmma_f32_16x16x4_f32(
                    false, a1, false, b, (short)0, acc[1][j], false, false);
            }
        }
        wait_async_copies();
        __syncthreads();
    }

    // Epilogue: C/D layout -> VGPR v holds row M = v + 8*half, col N = l16
#pragma unroll
    for (int s = 0; s < 2; ++s) {
#pragma unroll
        for (int j = 0; j < 4; ++j) {
            const int col = n0 + j * 16 + l16;
            const float bv = bias[col];
#pragma unroll
            for (int v = 0; v < 8; ++v) {
                const int row = m0 + wave * 32 + s * 16 + half * 8 + v;
                Y[(size_t)row * DIM + col] = acc[s][j][v] + bv;
            }
        }
    }
}

// ---------------------------------------------------------------------------
// Per-channel BatchNorm statistics (training mode, population variance).
// One block per channel; fold into scale/shift: y = x*scale + shift.
// ---------------------------------------------------------------------------
__global__ __launch_bounds__(256) void bn_stats(
    const float* __restrict__ Y, int R,
    const float* __restrict__ gamma, const float* __restrict__ beta,
    float* __restrict__ scale, float* __restrict__ shift)
{
    __shared__ float ssum[256];
    __shared__ float ssq[256];
    const int c = blockIdx.x;
    const int tid = threadIdx.x;
    float s = 0.f, q = 0.f;
    for (int r = tid; r < R; r += 256) {
        float v = Y[(size_t)r * DIM + c];
        s += v;
        q += v * v;
    }
    ssum[tid] = s;
    ssq[tid]  = q;
    __syncthreads();
    for (int off = 128; off > 0; off >>= 1) {
        if (tid < off) {
            ssum[tid] += ssum[tid + off];
            ssq[tid]  += ssq[tid + off];
        }
        __syncthreads();
    }
    if (tid == 0) {
        const float inv = 1.0f / (float)R;
        const float mu  = ssum[0] * inv;
        const float var = ssq[0] * inv - mu * mu;
        const float sc  = gamma[c] * rsqrtf(var + 1e-5f);
        scale[c] = sc;
        shift[c] = beta[c] - mu * sc;
    }
}

// ---------------------------------------------------------------------------
// In-place BN + ReLU (vectorized float4).
// ---------------------------------------------------------------------------
__global__ __launch_bounds__(256) void bn_relu_apply(
    float* __restrict__ Y, const float* __restrict__ sc,
    const float* __restrict__ sh, int n4)
{
    const int i = blockIdx.x * 256 + threadIdx.x;
    if (i >= n4) return;
    const int f = i & (DIM / 4 - 1);
    float4 v = reinterpret_cast<float4*>(Y)[i];
    const float4 s = reinterpret_cast<const float4*>(sc)[f];
    const float4 h = reinterpret_cast<const float4*>(sh)[f];
    v.x = fmaxf(v.x * s.x + h.x, 0.f);
    v.y = fmaxf(v.y * s.y + h.y, 0.f);
    v.z = fmaxf(v.z * s.z + h.z, 0.f);
    v.w = fmaxf(v.w * s.w + h.w, 0.f);
    reinterpret_cast<float4*>(Y)[i] = v;
}

// ---------------------------------------------------------------------------
// 3-NN inverse-distance interpolation of feats1 at xyz2 + BN2+ReLU(Y2) add.
// One wave (32 lanes) per destination point; 8 points per block.
// ---------------------------------------------------------------------------
__global__ __launch_bounds__(256) void interp_add(
    const float* __restrict__ xyz1, const float* __restrict__ xyz2,
    const float* __restrict__ feats1, const float* __restrict__ Y2,
    const float* __restrict__ scale2, const float* __restrict__ shift2,
    float* __restrict__ outF)
{
    __shared__ float sd[8][96];
    __shared__ int   si[8][96];
    __shared__ float sw[8][3];
    __shared__ int   sj[8][3];

    const int tid  = threadIdx.x;
    const int wave = tid >> 5;
    const int lane = tid & 31;
    const int p = blockIdx.x * 8 + wave;   // destination point id in [0, R2)
    const int b = p >> 12;                 // / N2C

    const float* xd = xyz2 + (size_t)p * 3;
    const float dx = xd[0], dy = xd[1], dz = xd[2];
    const float* xs = xyz1 + (size_t)b * N1C * 3;

    float d0 = 3.0e38f, d1 = 3.0e38f, d2 = 3.0e38f;
    int   i0 = 0, i1 = 0, i2 = 0;
    for (int s = lane; s < N1C; s += 32) {
        const float ax = xs[s * 3 + 0] - dx;
        const float ay = xs[s * 3 + 1] - dy;
        const float az = xs[s * 3 + 2] - dz;
        const float d  = ax * ax + ay * ay + az * az;
        if (d < d0)      { d2 = d1; i2 = i1; d1 = d0; i1 = i0; d0 = d; i0 = s; }
        else if (d < d1) { d2 = d1; i2 = i1; d1 = d;  i1 = s; }
        else if (d < d2) { d2 = d;  i2 = s; }
    }
    sd[wave][lane]      = d0; si[wave][lane]      = i0;
    sd[wave][lane + 32] = d1; si[wave][lane + 32] = i1;
    sd[wave][lane + 64] = d2; si[wave][lane + 64] = i2;
    __syncthreads();

    if (lane == 0) {
        float e0 = 3.0e38f, e1 = 3.0e38f, e2 = 3.0e38f;
        int   j0 = 0, j1 = 0, j2 = 0;
        for (int t = 0; t < 96; ++t) {
            const float d = sd[wave][t];
            const int   i = si[wave][t];
            if (d < e0)      { e2 = e1; j2 = j1; e1 = e0; j1 = j0; e0 = d; j0 = i; }
            else if (d < e1) { e2 = e1; j2 = j1; e1 = d;  j1 = i; }
            else if (d < e2) { e2 = d;  j2 = i; }
        }
        const float w0 = 1.f / (e0 + 1e-8f);
        const float w1 = 1.f / (e1 + 1e-8f);
        const float w2 = 1.f / (e2 + 1e-8f);
        const float wsum = w0 + w1 + w2;
        sw[wave][0] = w0 / wsum; sw[wave][1] = w1 / wsum; sw[wave][2] = w2 / wsum;
        sj[wave][0] = j0;        sj[wave][1] = j1;        sj[wave][2] = j2;
    }
    __syncthreads();

    const float w0 = sw[wave][0], w1 = sw[wave][1], w2 = sw[wave][2];
    const float4* f0 = (const float4*)(feats1 + ((size_t)b * N1C + sj[wave][0]) * DIM);
    const float4* f1 = (const float4*)(feats1 + ((size_t)b * N1C + sj[wave][1]) * DIM);
    const float4* f2 = (const float4*)(feats1 + ((size_t)b * N1C + sj[wave][2]) * DIM);
    const float4* y  = (const float4*)(Y2 + (size_t)p * DIM);
    const float4* sc = (const float4*)scale2;
    const float4* sh = (const float4*)shift2;
    float4* o = (float4*)(outF + (size_t)p * DIM);
#pragma unroll
    for (int q = 0; q < 4; ++q) {
        const int f = lane + 32 * q;          // float4 index 0..127 (coalesced)
        const float4 a0 = f0[f], a1 = f1[f], a2 = f2[f];
        const float4 yy = y[f], s4 = sc[f], h4 = sh[f];
        float4 r;
        r.x = w0 * a0.x + w1 * a1.x + w2 * a2.x + fmaxf(yy.x * s4.x + h4.x, 0.f);
        r.y = w0 * a0.y + w1 * a1.y + w2 * a2.y + fmaxf(yy.y * s4.y + h4.y, 0.f);
        r.z = w0 * a0.z + w1 * a1.z + w2 * a2.z + fmaxf(yy.z * s4.z + h4.z, 0.f);
        r.w = w0 * a0.w + w1 * a1.w + w2 * a2.w + fmaxf(yy.w * s4.w + h4.w, 0.f);
        o[f] = r;
    }
}

// ---------------------------------------------------------------------------
extern "C" void kernel_launch(void* const* d_in, const int* in_sizes, int n_in,
                              void* d_out, int out_size, void* d_ws, size_t ws_size,
                              hipStream_t stream)
{
    const float* xyz1    = (const float*)d_in[0];
    const float* points1 = (const float*)d_in[1];
    const float* xyz2    = (const float*)d_in[2];
    const float* points2 = (const float*)d_in[3];
    const float* fc1_w   = (const float*)d_in[4];
    const float* fc1_b   = (const float*)d_in[5];
    const float* g1      = (const float*)d_in[6];
    const float* be1     = (const float*)d_in[7];
    const float* fc2_w   = (const float*)d_in[8];
    const float* fc2_b   = (const float*)d_in[9];
    const float* g2      = (const float*)d_in[10];
    const float* be2     = (const float*)d_in[11];

    float* Y1     = (float*)d_ws;                      // [R1, DIM]  32 MB
    float* Y2     = Y1 + (size_t)R1 * DIM;             // [R2, DIM] 128 MB
    float* scale1 = Y2 + (size_t)R2 * DIM;
    float* shift1 = scale1 + DIM;
    float* scale2 = shift1 + DIM;
    float* shift2 = scale2 + DIM;

    float* out  = (float*)d_out;
    float* outF = out + (size_t)BC * N2C * 3;          // features after xyz2

    // Output tuple element 0: xyz2 passthrough
    (void)hipMemcpyAsync(out, xyz2, (size_t)BC * N2C * 3 * sizeof(float),
                         hipMemcpyDeviceToDevice, stream);

    gemm_bias_wmma<2 * DIM><<<dim3(R1 / TM, DIM / TN), NTHR, 0, stream>>>(points1, fc1_w, fc1_b, Y1);
    gemm_bias_wmma<DIM><<<dim3(R2 / TM, DIM / TN), NTHR, 0, stream>>>(points2, fc2_w, fc2_b, Y2);
    bn_stats<<<DIM, 256, 0, stream>>>(Y1, R1, g1, be1, scale1, shift1);
    bn_stats<<<DIM, 256, 0, stream>>>(Y2, R2, g2, be2, scale2, shift2);
    bn_relu_apply<<<(R1 * DIM / 4) / 256, 256, 0, stream>>>(Y1, scale1, shift1, R1 * DIM / 4);
    interp_add<<<R2 / 8, 256, 0, stream>>>(xyz1, xyz2, Y1, Y2, scale2, shift2, outF);
}